// CentroidHead_7473243095300
// MI455X (gfx1250) — compile-verified
//
#include <hip/hip_runtime.h>
#include <hip/hip_bf16.h>
#include <math.h>

#define NPTS      1000000
#define FIN       64
#define HID       32
#define SCORE_T   0.1f
#define CENT_T    0.2f
#define KOUT      128          // MAX_TREES * B
#define GB        2
#define GX        514
#define GY        514
#define GZ        66
#define GRID_CELLS ((long long)GB * GX * GY * GZ)   // 34,873,872 cells (~139.5 MB as u32)
#define PEAK_CAP  (1 << 18)

typedef __attribute__((ext_vector_type(2))) float v2f;
typedef __attribute__((ext_vector_type(8))) float v8f;

// ---------------------------------------------------------------------------
// K1: zero the scatter-max grid (uint encoding of non-negative f32 scores) and
// reset the peak-compaction counter.
// ---------------------------------------------------------------------------
__global__ void k_init(unsigned* __restrict__ grid, unsigned* __restrict__ counter,
                       long long cells) {
  long long i = (long long)blockIdx.x * blockDim.x + threadIdx.x;
  if (i == 0) *counter = 0u;
  long long stride = (long long)gridDim.x * blockDim.x;
  for (; i < cells; i += stride) grid[i] = 0u;
}

// ---------------------------------------------------------------------------
// K2: fused MLP (WMMA f32 16x16x4) + sigmoid + mask + grid scatter-max.
// One wave32 handles a 16-row tile of points. Block = 256 threads = 8 waves.
// ---------------------------------------------------------------------------
__global__ __launch_bounds__(256) void k_mlp(
    const float* __restrict__ feats,
    const int* __restrict__ cb, const int* __restrict__ cx,
    const int* __restrict__ cy, const int* __restrict__ cz,
    const unsigned char* __restrict__ mask,
    const float* __restrict__ W1, const float* __restrict__ b1,
    const float* __restrict__ W2, const float* __restrict__ b2,
    float* __restrict__ score_out, unsigned* __restrict__ grid) {
  // W1 transposed: sW1T[n][k] = W1[k][n]  ->  B-tile pair loads are contiguous b64
  __shared__ float sW1T[HID * FIN];
  __shared__ float sW2[HID];
  __shared__ float sB1[HID];

  const int t = threadIdx.x;
  for (int i = t; i < HID * FIN; i += 256) {
    int n = i >> 6;        // 0..31
    int k = i & 63;        // 0..63
    sW1T[i] = W1[k * HID + n];
  }
  if (t < HID) { sW2[t] = W2[t]; sB1[t] = b1[t]; }
  __syncthreads();

  const int wave = t >> 5;
  const int lane = t & 31;
  const int m  = lane & 15;   // row within tile (A) / column N (B,C,D)
  const int kh = lane >> 4;   // K half-select per ISA f32 16x4 layout

  const long long base = ((long long)blockIdx.x * 8 + wave) * 16;
  long long row  = base + m;
  long long lrow = row < NPTS ? row : (NPTS - 1);   // clamp tail (writes guarded)

  const float* ap  = feats + lrow * FIN + 2 * kh;           // A: feats[row][k0+2kh ..]
  const float* bp0 = &sW1T[m * FIN + 2 * kh];               // B tile, N = 0..15
  const float* bp1 = &sW1T[(m + 16) * FIN + 2 * kh];        // B tile, N = 16..31

  v8f acc0 = {};   // h[:, 0:16]
  v8f acc1 = {};   // h[:, 16:32]
#pragma unroll
  for (int k0 = 0; k0 < FIN; k0 += 4) {
    v2f a  = *(const v2f*)(ap  + k0);
    v2f bA = *(const v2f*)(bp0 + k0);
    v2f bB = *(const v2f*)(bp1 + k0);
    acc0 = __builtin_amdgcn_wmma_f32_16x16x4_f32(false, a, false, bA, (short)0, acc0, false, false);
    acc1 = __builtin_amdgcn_wmma_f32_16x16x4_f32(false, a, false, bB, (short)0, acc1, false, false);
  }

  // Layer 2: bias + relu + dot(h, W2) per row. C/D layout: lane holds column
  // N = m of rows M = v + 8*kh, so reduce across each 16-lane half.
  const float w2a = sW2[m], w2b = sW2[m + 16];
  const float bA1 = sB1[m], bB1 = sB1[m + 16];
  float part[8];
#pragma unroll
  for (int v = 0; v < 8; ++v) {
    float h0 = acc0[v] + bA1; h0 = h0 > 0.f ? h0 : 0.f;
    float h1 = acc1[v] + bB1; h1 = h1 > 0.f ? h1 : 0.f;
    part[v] = h0 * w2a + h1 * w2b;
  }
#pragma unroll
  for (int off = 1; off < 16; off <<= 1) {
#pragma unroll
    for (int v = 0; v < 8; ++v) part[v] += __shfl_xor(part[v], off, 32);
  }

  if (m == 0) {               // lanes 0 and 16 own rows base+0..7 / base+8..15
    const float bias2 = b2[0];
#pragma unroll
    for (int v = 0; v < 8; ++v) {
      long long r = base + v + 8 * kh;
      if (r < NPTS) {
        float o = part[v] + bias2;
        float s = 1.f / (1.f + __expf(-o));
        if (!mask[r]) s = 0.f;
        score_out[r] = s;
        if (s > SCORE_T) {
          int b = cb[r], x = cx[r] + 1, y = cy[r] + 1, z = cz[r] + 1;
          long long cell = ((((long long)b * GX + x) * GY + y) * GZ) + z;
          atomicMax(&grid[cell], __float_as_uint(s));  // u32 max == f32 max for s>0
        }
      }
    }
  }
}

// ---------------------------------------------------------------------------
// K3: 27-neighborhood local-max test + peak compaction.
// ---------------------------------------------------------------------------
__global__ __launch_bounds__(256) void k_peak(
    const float* __restrict__ score,
    const int* __restrict__ cb, const int* __restrict__ cx,
    const int* __restrict__ cy, const int* __restrict__ cz,
    const unsigned* __restrict__ grid,
    unsigned* __restrict__ counter,
    unsigned* __restrict__ pidx, float* __restrict__ pscore) {
  long long i = (long long)blockIdx.x * blockDim.x + threadIdx.x;
  if (i >= NPTS) return;
  float s = score[i];
  if (!(s > SCORE_T) || !(s > CENT_T)) return;   // cand && s>CENT_T prerequisite
  unsigned su = __float_as_uint(s);
  int b = cb[i], x = cx[i] + 1, y = cy[i] + 1, z = cz[i] + 1;
  unsigned mx = 0u;
#pragma unroll
  for (int dx = -1; dx <= 1; ++dx) {
#pragma unroll
    for (int dy = -1; dy <= 1; ++dy) {
      const unsigned* p = grid + ((((long long)b * GX + (x + dx)) * GY + (y + dy)) * GZ) + z;
      unsigned a = p[-1], c = p[0], d = p[1];
      mx = mx > a ? mx : a;
      mx = mx > c ? mx : c;
      mx = mx > d ? mx : d;
    }
  }
  if (mx == su) {   // hmax == s (bit-exact, same comparison the reference makes)
    unsigned pos = atomicAdd(counter, 1u);
    if (pos < PEAK_CAP) { pidx[pos] = (unsigned)i; pscore[pos] = s; }
  }
}

// ---------------------------------------------------------------------------
// K4: top-128 selection (lowest-index tie-break, matching top_k order) +
// coord gather. Single block.
// ---------------------------------------------------------------------------
__global__ __launch_bounds__(256) void k_topk(
    const unsigned* __restrict__ counter,
    const unsigned* __restrict__ pidx, float* __restrict__ pscore,
    const int* __restrict__ cb, const int* __restrict__ cx,
    const int* __restrict__ cy, const int* __restrict__ cz,
    int* __restrict__ out_idx, float* __restrict__ out_score,
    int* __restrict__ out_coords) {
  __shared__ float sv[256];
  __shared__ int   si[256];
  __shared__ int   sp[256];
  const int t = threadIdx.x;
  unsigned cnt = *counter;
  const int M = (int)(cnt < (unsigned)PEAK_CAP ? cnt : (unsigned)PEAK_CAP);

  for (int k = 0; k < KOUT; ++k) {
    float bs = -1.f; int bi = 0x7fffffff; int bp = -1;
    for (int j = t; j < M; j += 256) {
      float s = pscore[j];
      int idx = (int)pidx[j];
      if (s > bs || (s == bs && idx < bi)) { bs = s; bi = idx; bp = j; }
    }
    sv[t] = bs; si[t] = bi; sp[t] = bp;
    __syncthreads();
    for (int off = 128; off > 0; off >>= 1) {
      if (t < off) {
        if (sv[t + off] > sv[t] || (sv[t + off] == sv[t] && si[t + off] < si[t])) {
          sv[t] = sv[t + off]; si[t] = si[t + off]; sp[t] = sp[t + off];
        }
      }
      __syncthreads();
    }
    if (t == 0) {
      if (sp[0] >= 0 && sv[0] > 0.f) {
        int idx = si[0];
        out_idx[k]   = idx;
        out_score[k] = sv[0];
        out_coords[k * 4 + 0] = cb[idx];
        out_coords[k * 4 + 1] = cx[idx];
        out_coords[k * 4 + 2] = cy[idx];
        out_coords[k * 4 + 3] = cz[idx];
        pscore[sp[0]] = -2.f;            // exclude from later rounds
      } else {
        int idx = k;                      // filler (reference fills with -1.0)
        out_idx[k]   = idx;
        out_score[k] = -1.f;
        out_coords[k * 4 + 0] = cb[idx];
        out_coords[k * 4 + 1] = cx[idx];
        out_coords[k * 4 + 2] = cy[idx];
        out_coords[k * 4 + 3] = cz[idx];
      }
    }
    __syncthreads();
  }
}

// ---------------------------------------------------------------------------
extern "C" void kernel_launch(void* const* d_in, const int* in_sizes, int n_in,
                              void* d_out, int out_size, void* d_ws, size_t ws_size,
                              hipStream_t stream) {
  const float*         feats = (const float*)d_in[0];
  const int*           cb    = (const int*)d_in[1];
  const int*           cx    = (const int*)d_in[2];
  const int*           cy    = (const int*)d_in[3];
  const int*           cz    = (const int*)d_in[4];
  const unsigned char* mask  = (const unsigned char*)d_in[5];  // jax bool_ = 1 byte
  const float*         W1    = (const float*)d_in[6];
  const float*         b1    = (const float*)d_in[7];
  const float*         W2    = (const float*)d_in[8];
  const float*         b2    = (const float*)d_in[9];

  // Output layout (tuple concat): score[N] | peak_indices[128] | peak_scores[128] | peak_coords[128*4]
  float* score_out  = (float*)d_out;
  int*   out_idx    = (int*)d_out + NPTS;
  float* out_score  = (float*)d_out + NPTS + KOUT;
  int*   out_coords = (int*)d_out + NPTS + 2 * KOUT;

  // Workspace layout: grid (u32 cells) | counter | peak idx buf | peak score buf
  unsigned* grid    = (unsigned*)d_ws;
  unsigned* counter = grid + GRID_CELLS;
  unsigned* pidx    = counter + 64;                 // padding for alignment
  float*    pscore  = (float*)(pidx + PEAK_CAP);

  k_init<<<4096, 256, 0, stream>>>(grid, counter, GRID_CELLS);

  const int mlp_blocks = (NPTS + 127) / 128;        // 128 points per 256-thread block
  k_mlp<<<mlp_blocks, 256, 0, stream>>>(feats, cb, cx, cy, cz, mask,
                                        W1, b1, W2, b2, score_out, grid);

  k_peak<<<(NPTS + 255) / 256, 256, 0, stream>>>(score_out, cb, cx, cy, cz,
                                                 grid, counter, pidx, pscore);

  k_topk<<<1, 256, 0, stream>>>(counter, pidx, pscore, cb, cx, cy, cz,
                                out_idx, out_score, out_coords);
}